// SimpleLightGCN_24455543783537
// MI455X (gfx1250) — compile-verified
//
#include <hip/hip_runtime.h>

#define NUM_USERS 100000
#define NUM_ITEMS 50000
#define N_NODES   150000
#define EMB_DIM   64
#define N_LAYERS  3

typedef __attribute__((ext_vector_type(2))) float v2f;
typedef __attribute__((ext_vector_type(8))) float v8f;

// Guaranteed hardware fire-and-forget f32 atomic add (no CAS fallback).
__device__ __forceinline__ void hw_atomic_add_f32(float* addr, float val) {
    asm volatile("global_atomic_add_f32 %0, %1, off"
                 :
                 : "v"(addr), "v"(val)
                 : "memory");
}

// ---------------------------------------------------------------------------
// Kernel 1: cur = concat(user, item); acc = cur; nxt = 0   (float4 passes)
// ---------------------------------------------------------------------------
__global__ void lgcn_init(const float4* __restrict__ u, const float4* __restrict__ it,
                          float4* __restrict__ cur, float4* __restrict__ acc,
                          float4* __restrict__ nxt, int nU4, int nTot4)
{
    int i = blockIdx.x * blockDim.x + threadIdx.x;
    int stride = gridDim.x * blockDim.x;
    float4 z = make_float4(0.f, 0.f, 0.f, 0.f);
    for (; i < nTot4; i += stride) {
        float4 v = (i < nU4) ? u[i] : it[i - nU4];
        cur[i] = v;
        acc[i] = v;
        nxt[i] = z;
    }
}

// ---------------------------------------------------------------------------
// Kernel 2: COO SpMM scatter:  y[rows[e],:] += vals[e] * x[cols[e],:]
// One wave stages 32 edges (coalesced index loads), then broadcasts each
// edge via v_readlane (wave-uniform index -> scalar row/col/val); each of
// the 32 lanes handles 2 of the 64 columns with a float2 gather + two
// non-returning hardware f32 atomics (L2-resident working set).
// ---------------------------------------------------------------------------
__global__ void lgcn_spmm(const float* __restrict__ x, float* __restrict__ y,
                          const float* __restrict__ vals,
                          const int*   __restrict__ rows,
                          const int*   __restrict__ cols, int nEdges)
{
    const int lane   = threadIdx.x & 31;
    const int wave   = (blockIdx.x * blockDim.x + threadIdx.x) >> 5;
    const int nWaves = (gridDim.x * blockDim.x) >> 5;
    const int step   = nWaves * 32;

    for (int base = wave * 32; base < nEdges; base += step) {
        int myE = base + lane;
        // stream-ahead prefetch of the edge arrays (global_prefetch_b8)
        __builtin_prefetch(rows + myE + step, 0, 1);
        __builtin_prefetch(cols + myE + step, 0, 1);
        __builtin_prefetch(vals + myE + step, 0, 1);

        int r = 0, c = 0; int vbits = 0;
        if (myE < nEdges) {
            r = rows[myE];
            c = cols[myE];
            vbits = __float_as_int(vals[myE]);
        }

        int cnt = nEdges - base; if (cnt > 32) cnt = 32;
        for (int i = 0; i < cnt; ++i) {
            int   ri = __builtin_amdgcn_readlane(r, i);       // v_readlane_b32
            int   ci = __builtin_amdgcn_readlane(c, i);
            float vi = __int_as_float(__builtin_amdgcn_readlane(vbits, i));
            float2 xv = ((const float2*)(x + (size_t)ci * EMB_DIM))[lane];
            float* dst = y + (size_t)ri * EMB_DIM + lane * 2;
            hw_atomic_add_f32(dst,     vi * xv.x);
            hw_atomic_add_f32(dst + 1, vi * xv.y);
        }
    }
}

// ---------------------------------------------------------------------------
// Kernel 3: acc += nxt; zero the old cur buffer (becomes next layer's target)
// ---------------------------------------------------------------------------
__global__ void lgcn_accum(float4* __restrict__ acc, const float4* __restrict__ nxt,
                           float4* __restrict__ oldcur, int nTot4)
{
    int i = blockIdx.x * blockDim.x + threadIdx.x;
    int stride = gridDim.x * blockDim.x;
    float4 z = make_float4(0.f, 0.f, 0.f, 0.f);
    for (; i < nTot4; i += stride) {
        float4 a = acc[i];
        float4 b = nxt[i];
        a.x += b.x; a.y += b.y; a.z += b.z; a.w += b.w;
        acc[i] = a;
        oldcur[i] = z;
    }
}

// ---------------------------------------------------------------------------
// Kernel 4: final scale acc *= 0.25, done with V_WMMA_F32_16X16X4_F32.
// Per 16x16 tile X:  D = sum_j A_j * B_j, with A_j = 0.25 * (16x4 block j of
// I16) and B_j = rows 4j..4j+3 of X  =>  D = 0.25 * X exactly (f32 FMA path).
// A layout (16x4, 2 VGPRs): lanes 0-15 -> M=lane,(K=0,K=1); lanes 16-31 ->
// M=lane-16,(K=2,K=3).  B layout (4x16, 2 VGPRs): lanes 0-15 -> N=lane,
// (K=0,K=1); lanes 16-31 -> N=lane-16,(K=2,K=3).
// One wave per tile; wave-uniform guard keeps EXEC all-ones around WMMA.
// ---------------------------------------------------------------------------
__global__ void lgcn_scale_wmma(float* __restrict__ acc, int nTiles)
{
    const int lane = threadIdx.x & 31;
    const int wave = (blockIdx.x * blockDim.x + threadIdx.x) >> 5;
    if (wave >= nTiles) return;            // whole-wave exit, EXEC stays full

    const int rowTile = wave >> 2;         // 9375 row tiles
    const int colTile = wave & 3;          // 4 col tiles of 16
    float* base = acc + (size_t)rowTile * 16 * EMB_DIM + colTile * 16;

    const int half = lane >> 4;            // 0: lanes 0-15, 1: lanes 16-31
    const int n    = lane & 15;            // column within tile / M for A

    v8f d = {};                            // C starts at 0, accumulates
#pragma unroll
    for (int j = 0; j < 4; ++j) {
        const int kb = 4 * j + (half ? 2 : 0);     // B rows for this lane half
        v2f b;
        b.x = base[(kb + 0) * EMB_DIM + n];
        b.y = base[(kb + 1) * EMB_DIM + n];
        v2f a;                                     // 0.25 * identity block j
        a.x = (n == kb + 0) ? 0.25f : 0.0f;
        a.y = (n == kb + 1) ? 0.25f : 0.0f;
        d = __builtin_amdgcn_wmma_f32_16x16x4_f32(
                /*neg_a=*/false, a, /*neg_b=*/false, b,
                /*c_mod=*/(short)0, d, /*reuse_a=*/false, /*reuse_b=*/false);
    }
    // D layout: VGPR v -> row v (lanes 0-15) / row v+8 (lanes 16-31), col n
#pragma unroll
    for (int v = 0; v < 8; ++v) {
        base[(v + half * 8) * EMB_DIM + n] = d[v];
    }
}

// ---------------------------------------------------------------------------
extern "C" void kernel_launch(void* const* d_in, const int* in_sizes, int n_in,
                              void* d_out, int out_size, void* d_ws, size_t ws_size,
                              hipStream_t stream) {
    const float* user_emb = (const float*)d_in[0];
    const float* item_emb = (const float*)d_in[1];
    const float* adj_vals = (const float*)d_in[2];
    const int*   adj_rows = (const int*)d_in[3];
    const int*   adj_cols = (const int*)d_in[4];
    const int nEdges = in_sizes[2];

    float* acc = (float*)d_out;                         // 150000 x 64
    float* cur = (float*)d_ws;                          // ping
    float* nxt = cur + (size_t)N_NODES * EMB_DIM;       // pong

    const int nTot4 = N_NODES * EMB_DIM / 4;            // 2.4M float4
    const int nU4   = NUM_USERS * EMB_DIM / 4;

    lgcn_init<<<2048, 256, 0, stream>>>((const float4*)user_emb, (const float4*)item_emb,
                                        (float4*)cur, (float4*)acc, (float4*)nxt,
                                        nU4, nTot4);

    for (int l = 0; l < N_LAYERS; ++l) {
        lgcn_spmm<<<4096, 256, 0, stream>>>(cur, nxt, adj_vals, adj_rows, adj_cols, nEdges);
        lgcn_accum<<<2048, 256, 0, stream>>>((float4*)acc, (const float4*)nxt,
                                             (float4*)cur, nTot4);
        float* t = cur; cur = nxt; nxt = t;             // swap ping/pong
    }

    const int nTiles = (N_NODES / 16) * (EMB_DIM / 16); // 37500 tiles, 1 wave each
    const int blocks = (nTiles * 32 + 255) / 256;
    lgcn_scale_wmma<<<blocks, 256, 0, stream>>>(acc, nTiles);
}